// attention_pooling_16716012716597
// MI455X (gfx1250) — compile-verified
//
#include <hip/hip_runtime.h>
#include <hip/hip_bf16.h>

#ifndef __has_builtin
#define __has_builtin(x) 0
#endif

#if __has_builtin(__builtin_amdgcn_tensor_load_to_lds)
#define USE_TDM 1
#else
#define USE_TDM 0
#endif

typedef float v2f  __attribute__((ext_vector_type(2)));
typedef float v8f  __attribute__((ext_vector_type(8)));
typedef unsigned int u32x4 __attribute__((ext_vector_type(4)));
typedef int   i32x4 __attribute__((ext_vector_type(4)));
typedef int   i32x8 __attribute__((ext_vector_type(8)));

constexpr int B = 32, S = 4096, D = 2048, R = 128;
constexpr float INV_SQRT_R = 0.08838834764831845f;  // 1/sqrt(128)

// LDS tile: 32 rows x 2048 f32, +1 DWORD pad every 256 DWORDs (matches TDM
// pad_interval=256dw / pad_amount=1dw) -> row stride 2056 words => 8-bank skew.
constexpr int TILE_ELEMS = 32 * D;                   // 65536
constexpr int TILE_PADDED = TILE_ELEMS + (TILE_ELEMS >> 8); // 65792
constexpr int SMEM_FLOATS = TILE_PADDED + D + 128;   // tile + v + partials
constexpr int SMEM_BYTES  = SMEM_FLOATS * 4;         // 271,872 B < 320 KB/WGP

__device__ __forceinline__ int padidx(int e) { return e + (e >> 8); }

// ---------------------------------------------------------------- kernel 1
// qq = query@Wq + bq ; v = (We@qq)/sqrt(R) ; sbias = (be.qq)/sqrt(R)
__global__ __launch_bounds__(256) void proj_kernel(
    const float* __restrict__ q, const float* __restrict__ Wq,
    const float* __restrict__ bq, const float* __restrict__ We,
    const float* __restrict__ be, float* __restrict__ v_out,
    float* __restrict__ sb_out) {
  __shared__ float qq[R];
  const int tid = threadIdx.x;
  if (tid < R) {
    float acc = bq[tid];
    for (int d = 0; d < D; ++d) acc = fmaf(q[d], Wq[(size_t)d * R + tid], acc);
    qq[tid] = acc;
  }
  __syncthreads();
  for (int d = tid; d < D; d += 256) {
    float acc = 0.f;
    for (int r = 0; r < R; ++r) acc = fmaf(We[(size_t)d * R + r], qq[r], acc);
    v_out[d] = acc * INV_SQRT_R;
  }
  if (tid == 0) {
    float sbv = 0.f;
    for (int r = 0; r < R; ++r) sbv = fmaf(be[r], qq[r], sbv);
    sb_out[0] = sbv * INV_SQRT_R;
  }
}

// ---------------------------------------------------------------- kernel 2
// Per WG: TDM-load a 32x2048 f32 tile to LDS, then f32 WMMA dot against v.
__global__ __launch_bounds__(256) void score_kernel(
    const float* __restrict__ x, const float* __restrict__ v,
    const float* __restrict__ sb, float* __restrict__ scores) {
  extern __shared__ float smem[];
  float* tile = smem;                       // TILE_PADDED floats
  float* vlds = smem + TILE_PADDED;         // D floats
  float* part = smem + TILE_PADDED + D;     // 8 waves x 16 token-partials

  const int tid  = threadIdx.x;
  const int wave = tid >> 5;
  const int lane = tid & 31;
  const int b  = blockIdx.x >> 7;           // S/32 = 128 chunks per batch
  const int s0 = (blockIdx.x & 127) * 32;
  const size_t rowbase = ((size_t)b * S + s0) * (size_t)D;

#if USE_TDM
  if (wave == 0) {
    unsigned long long ga = (unsigned long long)(const void*)(x + rowbase);
    unsigned int lds_off  = (unsigned int)(unsigned long long)(void*)tile;
    u32x4 g0;
    g0[0] = 1u;                                           // count=1 (valid D#)
    g0[1] = lds_off;                                      // lds_addr
    g0[2] = (unsigned int)(ga & 0xFFFFFFFFu);             // global_addr[31:0]
    g0[3] = (unsigned int)((ga >> 32) & 0x01FFFFFFu)      // global_addr[56:32]
          | (2u << 30);                                   // type = 2 (image)
    i32x8 g1;
    g1[0] = (int)((2u << 16) | (1u << 20) | (7u << 22));  // 4B elems, pad 1dw/256dw
    g1[1] = (int)((unsigned)D << 16);                     // tensor_dim0 = 2048
    g1[2] = (int)(32u << 16);                             // tensor_dim1 = 32
    g1[3] = (int)((unsigned)D << 16);                     // tile_dim0   = 2048
    g1[4] = 32;                                           // tile_dim1   = 32
    g1[5] = D;                                            // dim0_stride = 2048
    g1[6] = 0;
    g1[7] = 0;
    i32x4 z4 = {0, 0, 0, 0};
#if defined(__clang_major__) && (__clang_major__ >= 23)
    i32x8 z8 = {0, 0, 0, 0, 0, 0, 0, 0};
    __builtin_amdgcn_tensor_load_to_lds(g0, g1, z4, z4, z8, 0);
#else
    __builtin_amdgcn_tensor_load_to_lds(g0, g1, z4, z4, 0);
#endif
  }
#endif
  for (int i = tid; i < D; i += 256) vlds[i] = v[i];
#if USE_TDM
  if (wave == 0) __builtin_amdgcn_s_wait_tensorcnt(0);
#else
  {
    const float4* src = (const float4*)(x + rowbase);
    for (int e4 = tid; e4 < TILE_ELEMS / 4; e4 += 256) {
      float4 val = src[e4];
      int o = padidx(e4 * 4);                // 4 elems never straddle a pad
      tile[o] = val.x; tile[o + 1] = val.y; tile[o + 2] = val.z; tile[o + 3] = val.w;
    }
  }
#endif
  __syncthreads();

  // wave -> (token group, d-quarter); 128 accumulating f32 WMMAs per wave.
  const int group   = wave >> 2;
  const int quarter = wave & 3;
  const int M  = lane & 15;
  const int hi = (lane >> 4) & 1;            // lanes 16-31 hold K=2,3
  const int token = group * 16 + M;
  const int dq = quarter * 512;

  v8f c = {};
  for (int k = 0; k < 128; ++k) {
    const int d0 = dq + k * 4;
    const int ea = token * D + d0 + (hi ? 2 : 0);
    v2f a, bb;
    a.x  = tile[padidx(ea)];
    a.y  = tile[padidx(ea + 1)];
    bb.x = vlds[d0 + (hi ? 1 : 0)];          // B[k][j] = v[d0+k] (all 16 cols)
    bb.y = vlds[d0 + (hi ? 3 : 2)];
    c = __builtin_amdgcn_wmma_f32_16x16x4_f32(false, a, false, bb,
                                              (short)0, c, false, false);
  }
  // Column 0 of D: lane 0 holds rows M=0..7, lane 16 holds rows M=8..15.
  if (lane == 0) {
#pragma unroll
    for (int i = 0; i < 8; ++i) part[wave * 16 + i] = c[i];
  } else if (lane == 16) {
#pragma unroll
    for (int i = 0; i < 8; ++i) part[wave * 16 + 8 + i] = c[i];
  }
  __syncthreads();
  if (tid < 32) {
    const int g = tid >> 4, lo = tid & 15;
    float ssum = part[(g * 4 + 0) * 16 + lo] + part[(g * 4 + 1) * 16 + lo] +
                 part[(g * 4 + 2) * 16 + lo] + part[(g * 4 + 3) * 16 + lo];
    scores[(size_t)b * S + s0 + tid] = ssum + sb[0];
  }
}

// ---------------------------------------------------------------- kernel 3
__global__ __launch_bounds__(256) void softmax_kernel(float* __restrict__ scores) {
  __shared__ float red[256];
  const int b = blockIdx.x, tid = threadIdx.x;
  float* sc = scores + (size_t)b * S;
  float m = -3.402823466e38f;
  for (int i = 0; i < S / 256; ++i) m = fmaxf(m, sc[tid + i * 256]);
  red[tid] = m; __syncthreads();
  for (int s = 128; s > 0; s >>= 1) {
    if (tid < s) red[tid] = fmaxf(red[tid], red[tid + s]);
    __syncthreads();
  }
  m = red[0]; __syncthreads();
  float l = 0.f;
  for (int i = 0; i < S / 256; ++i) l += expf(sc[tid + i * 256] - m);
  red[tid] = l; __syncthreads();
  for (int s = 128; s > 0; s >>= 1) {
    if (tid < s) red[tid] += red[tid + s];
    __syncthreads();
  }
  const float invl = 1.0f / red[0]; __syncthreads();
  for (int i = 0; i < S / 256; ++i) {
    const int idx = tid + i * 256;
    sc[idx] = expf(sc[idx] - m) * invl;
  }
}

// ---------------------------------------------------------------- kernel 4
// part[b, schunk, d] = sum_{s in chunk} attn[b,s] * x[b,s,d]
__global__ __launch_bounds__(256) void wsum_kernel(
    const float* __restrict__ x, const float* __restrict__ attn,
    float* __restrict__ part) {
  const int tid = threadIdx.x;
  const int d  = blockIdx.x * 256 + tid;   // 8 d-blocks
  const int sc = blockIdx.y;               // 8 s-chunks of 512
  const int b  = blockIdx.z;
  const int s0 = sc * 512;
  const float* xp = x + ((size_t)b * S + s0) * (size_t)D + d;
  const float* ap = attn + (size_t)b * S + s0;
  float acc = 0.f;
  for (int s = 0; s < 512; ++s) {
    if (s + 16 < 512) __builtin_prefetch(xp + (size_t)(s + 16) * D, 0, 0);
    acc = fmaf(ap[s], xp[(size_t)s * D], acc);
  }
  part[((size_t)(b * 8 + sc)) * D + d] = acc;
}

// ---------------------------------------------------------------- kernel 5
__global__ __launch_bounds__(256) void ln_kernel(
    const float* __restrict__ part, const float* __restrict__ gamma,
    const float* __restrict__ beta, float* __restrict__ out) {
  __shared__ float red[256];
  const int b = blockIdx.x, tid = threadIdx.x;
  float g[8];
  float lsum = 0.f;
#pragma unroll
  for (int j = 0; j < 8; ++j) {
    const int d = j * 256 + tid;
    float acc = 0.f;
    for (int c = 0; c < 8; ++c) acc += part[((size_t)(b * 8 + c)) * D + d];
    g[j] = acc; lsum += acc;
  }
  red[tid] = lsum; __syncthreads();
  for (int s = 128; s > 0; s >>= 1) {
    if (tid < s) red[tid] += red[tid + s];
    __syncthreads();
  }
  const float mean = red[0] * (1.0f / D); __syncthreads();
  float lvar = 0.f;
#pragma unroll
  for (int j = 0; j < 8; ++j) { const float t = g[j] - mean; lvar += t * t; }
  red[tid] = lvar; __syncthreads();
  for (int s = 128; s > 0; s >>= 1) {
    if (tid < s) red[tid] += red[tid + s];
    __syncthreads();
  }
  const float inv = rsqrtf(red[0] * (1.0f / D) + 1e-5f);
#pragma unroll
  for (int j = 0; j < 8; ++j) {
    const int d = j * 256 + tid;
    out[(size_t)b * D + d] = (g[j] - mean) * inv * gamma[d] + beta[d];
  }
}

// ---------------------------------------------------------------- launcher
extern "C" void kernel_launch(void* const* d_in, const int* in_sizes, int n_in,
                              void* d_out, int out_size, void* d_ws,
                              size_t ws_size, hipStream_t stream) {
  (void)in_sizes; (void)n_in; (void)out_size; (void)ws_size;
  const float* x     = (const float*)d_in[0];
  const float* query = (const float*)d_in[1];
  const float* Wq    = (const float*)d_in[2];
  const float* bq    = (const float*)d_in[3];
  const float* We    = (const float*)d_in[4];
  const float* be    = (const float*)d_in[5];
  const float* gamma = (const float*)d_in[6];
  const float* beta  = (const float*)d_in[7];
  float* out = (float*)d_out;
  float* ws  = (float*)d_ws;

  float* ws_v      = ws;                       // D
  float* ws_sb     = ws + D;                   // 1
  float* ws_scores = ws + 4096;                // B*S
  float* ws_part   = ws + 4096 + (size_t)B * S; // B*8*D

  hipFuncSetAttribute(reinterpret_cast<const void*>(score_kernel),
                      hipFuncAttributeMaxDynamicSharedMemorySize, SMEM_BYTES);

  proj_kernel<<<1, 256, 0, stream>>>(query, Wq, bq, We, be, ws_v, ws_sb);
  score_kernel<<<dim3(B * (S / 32)), 256, SMEM_BYTES, stream>>>(x, ws_v, ws_sb,
                                                                ws_scores);
  softmax_kernel<<<B, 256, 0, stream>>>(ws_scores);
  wsum_kernel<<<dim3(8, 8, B), 256, 0, stream>>>(x, ws_scores, ws_part);
  ln_kernel<<<B, 256, 0, stream>>>(ws_part, gamma, beta, out);
}